// PointPillarScatter_2860448219530
// MI455X (gfx1250) — compile-verified
//
#include <hip/hip_runtime.h>
#include <hip/hip_bf16.h>

// PointPillarScatter for gfx1250 (MI455X).
// Strategy: invert the scatter into a gather.
//   Pass 1: map[b*S + y*NX + x] = -1 for all cells          (3.4 MB, lives in L2)
//   Pass 2: atomicMax(map[cell], pillar_idx)                (deterministic "last wins")
//   Pass 3: out[b][c][s] = map[b*S+s] >= 0 ? pf[p][c] : 0   (single fully-coalesced
//           write pass over the 219 MB output, nontemporal b128 stores)

typedef float v4f __attribute__((ext_vector_type(4)));
typedef int   v4i __attribute__((ext_vector_type(4)));

constexpr int kNX = 432;
constexpr int kNY = 496;
constexpr int kC  = 64;
constexpr int kS  = kNX * kNY;   // 214272 spatial cells per batch (divisible by 4)

// ---------------------------------------------------------------------------
// Pass 1: fill cell->pillar map with -1 (vectorized int4 stores).
// ---------------------------------------------------------------------------
__global__ void pps_init_map(int* __restrict__ map, int n4) {
    int g = blockIdx.x * blockDim.x + threadIdx.x;
    if (g < n4) {
        ((v4i*)map)[g] = (v4i)(-1);
    }
}

// ---------------------------------------------------------------------------
// Pass 2: scatter pillar indices. atomicMax => largest pillar index wins,
// which matches "last update wins" (deterministic across replays).
// coords layout per pillar: (b, z, y, x), z == 0.
// ---------------------------------------------------------------------------
__global__ void pps_scatter_idx(const int* __restrict__ vc,
                                int* __restrict__ map, int n_pillars) {
    int p = blockIdx.x * blockDim.x + threadIdx.x;
    if (p >= n_pillars) return;
    int b = vc[p * 4 + 0];
    int y = vc[p * 4 + 2];
    int x = vc[p * 4 + 3];
    int flat = b * kS + y * kNX + x;
    atomicMax(&map[flat], p);
}

// ---------------------------------------------------------------------------
// Pass 3: gather. Each thread owns 4 consecutive spatial cells of one batch
// and produces all 64 channel values for them. Output writes are b128 and
// coalesced across the wave (wave32 * 16B = 512B contiguous per store op).
// Pillar rows (64 f32 = 256B contiguous) are read as float4 and transposed
// in registers. Empty cell-quads take a pure zero-store fast path.
// ---------------------------------------------------------------------------
__global__ void pps_gather(const float* __restrict__ pf,
                           const int* __restrict__ map,
                           float* __restrict__ out, int n_groups) {
    int g = blockIdx.x * blockDim.x + threadIdx.x;
    if (g >= n_groups) return;

    const int S4 = kS / 4;
    int b  = g / S4;
    int s4 = g - b * S4;
    int s  = s4 * 4;

    v4i m = ((const v4i*)map)[g];
    float* out_base = out + (size_t)b * kC * kS + s;

    int mmax = max(max(m.x, m.y), max(m.z, m.w));
    if (mmax < 0) {
        // All four cells empty: stream zeros, never pollute caches.
        v4f zv = (v4f)0.0f;
#pragma unroll 8
        for (int c = 0; c < kC; ++c) {
            __builtin_nontemporal_store(zv, (v4f*)(out_base + (size_t)c * kS));
        }
        return;
    }

    // Cover map->feature dependent-load latency (lowers to global_prefetch_b8).
    if (m.x >= 0) __builtin_prefetch(pf + (size_t)m.x * kC, 0, 0);
    if (m.y >= 0) __builtin_prefetch(pf + (size_t)m.y * kC, 0, 0);
    if (m.z >= 0) __builtin_prefetch(pf + (size_t)m.z * kC, 0, 0);
    if (m.w >= 0) __builtin_prefetch(pf + (size_t)m.w * kC, 0, 0);

    const v4f* pf4 = (const v4f*)pf;  // 16 float4 per pillar row
    const v4f zv = (v4f)0.0f;

    for (int cb = 0; cb < kC / 4; ++cb) {
        // One float4 (4 channels) per owned pillar.
        v4f r0 = (m.x >= 0) ? pf4[(size_t)m.x * 16 + cb] : zv;
        v4f r1 = (m.y >= 0) ? pf4[(size_t)m.y * 16 + cb] : zv;
        v4f r2 = (m.z >= 0) ? pf4[(size_t)m.z * 16 + cb] : zv;
        v4f r3 = (m.w >= 0) ? pf4[(size_t)m.w * 16 + cb] : zv;

        // Transpose 4x4 in registers: channel-major vectors over the 4 cells.
        v4f o0 = {r0.x, r1.x, r2.x, r3.x};
        v4f o1 = {r0.y, r1.y, r2.y, r3.y};
        v4f o2 = {r0.z, r1.z, r2.z, r3.z};
        v4f o3 = {r0.w, r1.w, r2.w, r3.w};

        float* p0 = out_base + (size_t)(cb * 4 + 0) * kS;
        __builtin_nontemporal_store(o0, (v4f*)p0);
        __builtin_nontemporal_store(o1, (v4f*)(p0 + (size_t)kS));
        __builtin_nontemporal_store(o2, (v4f*)(p0 + (size_t)2 * kS));
        __builtin_nontemporal_store(o3, (v4f*)(p0 + (size_t)3 * kS));
    }
}

extern "C" void kernel_launch(void* const* d_in, const int* in_sizes, int n_in,
                              void* d_out, int out_size, void* d_ws, size_t ws_size,
                              hipStream_t stream) {
    const float* pf = (const float*)d_in[0];   // [n_pillars, 64] f32
    const int*   vc = (const int*)d_in[1];     // [n_pillars, 4]  int

    float* out = (float*)d_out;                // [B, 64, 496, 432] f32
    int n_pillars = in_sizes[0] / kC;
    int B = out_size / (kC * kS);              // = 4

    int* map  = (int*)d_ws;                    // B*kS int32 (3.4 MB) in workspace
    int  mapN = B * kS;
    int  n4   = mapN / 4;

    pps_init_map<<<(n4 + 255) / 256, 256, 0, stream>>>(map, n4);
    pps_scatter_idx<<<(n_pillars + 255) / 256, 256, 0, stream>>>(vc, map, n_pillars);
    pps_gather<<<(n4 + 255) / 256, 256, 0, stream>>>(pf, map, out, n4);
}